// DeepFactorizationMachine_26156350832969
// MI455X (gfx1250) — compile-verified
//
#include <hip/hip_runtime.h>
#include <math.h>

#define B_SZ   2048
#define F_SZ   26
#define C_SZ   64
#define E_SZ   64
#define D_SZ   13
#define KTOT   (F_SZ * C_SZ)          // 1664
#define SPCOLS (F_SZ * (C_SZ + 1) - 1) // 1689

typedef float v2f __attribute__((ext_vector_type(2)));
typedef float v8f __attribute__((ext_vector_type(8)));

// ---------------------------------------------------------------------------
// Kernel 1: per-(batch,field) 64-bin histogram of sparse indices + linear term.
// Output HT is K-major: HT[(f*64 + r) * B_SZ + b]  (WMMA-A-friendly, coalesced)
// ---------------------------------------------------------------------------
__global__ void __launch_bounds__(128)
dfm_hist_kernel(const int* __restrict__ sp,        // [B][1689] indices in [0,64)
                const float* __restrict__ lin_emb, // [F*64]
                float* __restrict__ HT,            // [KTOT][B]
                float* __restrict__ linwx)         // [B][F]
{
    // 128 threads * 64 bins, pad stride 65 -> bank = (tx + r) % 64, conflict-free
    __shared__ float cnt[128 * 65];
    const int tx = threadIdx.x;
    const int b  = blockIdx.x * 128 + tx;
    const int f  = blockIdx.y;

    float* my = &cnt[tx * 65];
#pragma unroll
    for (int r = 0; r < 64; ++r) my[r] = 0.0f;

    const int*   row = sp + (size_t)b * SPCOLS + f * (C_SZ + 1);
    const float* lf  = lin_emb + f * C_SZ;
    float ls = 0.0f;
#pragma unroll 4
    for (int c = 0; c < 64; ++c) {
        int r = row[c];          // 0..63
        my[r] += 1.0f;           // private LDS region: no races
        ls    += lf[r];          // tiny table, L1-resident
    }
    linwx[b * F_SZ + f] = ls;

    const int kbase = f * C_SZ;
#pragma unroll 4
    for (int r = 0; r < 64; ++r) {
        // 128 consecutive b across threads -> fully coalesced 512B stores
        HT[(size_t)(kbase + r) * B_SZ + b] = my[r];
    }
}

// ---------------------------------------------------------------------------
// Kernel 2: s = H @ W, sq = H @ (W*W) via V_WMMA_F32_16X16X4_F32, then
// pairwise reduction + linear term + sigmoid.
// Block: 128 threads = 4 waves; wave w owns E-columns [16w, 16w+16).
// Grid: 128 blocks, block bx owns batch rows [16*bx, 16*bx+16).
// ---------------------------------------------------------------------------
__global__ void __launch_bounds__(128)
dfm_fm_kernel(const float* __restrict__ HT,    // [KTOT][B]
              const float* __restrict__ emb,   // [KTOT][E]  (= emb_tab flat)
              const float* __restrict__ linwx, // [B][F]
              const float* __restrict__ dense, // [B][D]
              const float* __restrict__ linW,  // [F+D]
              const float* __restrict__ linb,  // [1]
              float* __restrict__ out)         // [B]
{
    __shared__ float red[16];

    const int lane = threadIdx.x & 31;
    const int wv   = threadIdx.x >> 5;
    const int b0   = blockIdx.x * 16;
    const int m    = lane & 15;        // M (A) / N (B) position for this lane
    const int kh   = (lane >> 4) * 2;  // K sub-offset: lanes 0-15 -> {0,1}, 16-31 -> {2,3}
    const int e0   = wv * 16;

    if (threadIdx.x < 16) red[threadIdx.x] = 0.0f;
    __syncthreads();

    v8f accS = {};
    v8f accQ = {};

    // Per-lane base pointers (CDNA5 ISA 7.12.2 layouts, wave32):
    //  A 16x4 f32: VGPR0 = A[m][kh], VGPR1 = A[m][kh+1]
    //  B 4x16 f32: VGPR0 = B[kh][n], VGPR1 = B[kh+1][n]
    const float* Acol = HT  + b0 + m;   // columns of HT, lanes 0..15 contiguous
    const float* Brow = emb + e0 + m;   // rows of emb,   lanes 0..15 contiguous

#pragma unroll 2
    for (int k = 0; k < KTOT; k += 4) {
        v2f a, b, bq;
        a.x  = Acol[(size_t)(k + kh)     * B_SZ];
        a.y  = Acol[(size_t)(k + kh + 1) * B_SZ];
        b.x  = Brow[(size_t)(k + kh)     * E_SZ];
        b.y  = Brow[(size_t)(k + kh + 1) * E_SZ];
        bq.x = b.x * b.x;
        bq.y = b.y * b.y;
        // D = A*B + C : 8 args (neg_a, A, neg_b, B, c_mod, C, reuse_a, reuse_b)
        accS = __builtin_amdgcn_wmma_f32_16x16x4_f32(
                   false, a, false, b,  (short)0, accS, false, false);
        accQ = __builtin_amdgcn_wmma_f32_16x16x4_f32(
                   false, a, false, bq, (short)0, accQ, false, false);
    }

    // C/D layout: VGPR i -> row m = i + (lane>=16 ? 8 : 0), col = lane%16 + e0.
    // fm_pairwise needs sum over all E columns of (s^2 - sq) per row.
    const int mbase = (lane >> 4) * 8;
#pragma unroll
    for (int i = 0; i < 8; ++i) {
        float p = accS[i] * accS[i] - accQ[i];
        atomicAdd(&red[mbase + i], p);   // ds_add_f32
    }
    __syncthreads();

    if (threadIdx.x < 16) {
        const int brow = b0 + threadIdx.x;
        float pair = 0.5f * red[threadIdx.x];
        float lin  = linb[0];
        const float* lw = linwx + brow * F_SZ;
#pragma unroll
        for (int f = 0; f < F_SZ; ++f) lin += lw[f] * linW[f];
        const float* dd = dense + brow * D_SZ;
#pragma unroll
        for (int d = 0; d < D_SZ; ++d) lin += dd[d] * linW[F_SZ + d];
        float x = lin + pair;
        out[brow] = 1.0f / (1.0f + expf(-x));
    }
}

// ---------------------------------------------------------------------------
// Host launcher
// ---------------------------------------------------------------------------
extern "C" void kernel_launch(void* const* d_in, const int* in_sizes, int n_in,
                              void* d_out, int out_size, void* d_ws, size_t ws_size,
                              hipStream_t stream)
{
    (void)in_sizes; (void)n_in; (void)out_size; (void)ws_size;

    const int*   sparse = (const int*)  d_in[0]; // [B][1689]
    const float* dense  = (const float*)d_in[1]; // [B][13]
    const float* linemb = (const float*)d_in[2]; // [26][64][1]
    const float* embtab = (const float*)d_in[3]; // [26][64][64] == [KTOT][E]
    const float* linW   = (const float*)d_in[4]; // [39][1]
    const float* linb   = (const float*)d_in[5]; // [1]
    float*       outp   = (float*)d_out;         // [B]

    float* HT    = (float*)d_ws;                         // KTOT * B floats (13.6 MB)
    float* linwx = HT + (size_t)KTOT * B_SZ;             // B * F floats

    dim3 g1(B_SZ / 128, F_SZ);
    dfm_hist_kernel<<<g1, 128, 0, stream>>>(sparse, linemb, HT, linwx);

    dfm_fm_kernel<<<B_SZ / 16, 128, 0, stream>>>(HT, embtab, linwx, dense,
                                                 linW, linb, outp);
}